// Conv_agg_raw_28527172780755
// MI455X (gfx1250) — compile-verified
//
#include <hip/hip_runtime.h>
#include <cstdint>

typedef __attribute__((ext_vector_type(2))) float v2f;
typedef __attribute__((ext_vector_type(8))) float v8f;

static constexpr int NN = 4096;   // nodes
static constexpr int KK = 8;      // adjacency channels
static constexpr int CC = 256;    // channels
static constexpr int EE = 131072; // edges
static constexpr int KTOT = KK * CC;        // 2048 contraction depth
static constexpr int KCHUNK = 512;          // K staged in LDS per step
static constexpr int LDSTRIDE = KCHUNK + 4; // 516: 516 % 64 == 4 -> conflict-free frag loads

// ---------------- CSR-by-dst build ----------------
__global__ void k_zero(int* __restrict__ cnt, int* __restrict__ cur) {
    int i = blockIdx.x * 256 + threadIdx.x;
    if (i < NN) { cnt[i] = 0; cur[i] = 0; }
}

__global__ void k_hist(const int* __restrict__ ei, int* __restrict__ cnt) {
    int e = blockIdx.x * 256 + threadIdx.x;
    if (e < EE) atomicAdd(&cnt[ei[EE + e]], 1);
}

__global__ void k_scan(const int* __restrict__ cnt, int* __restrict__ offs) {
    if (threadIdx.x == 0) {
        int s = 0;
        for (int i = 0; i < NN; ++i) { offs[i] = s; s += cnt[i]; }
        offs[NN] = s;
    }
}

__global__ void k_scatter(const int* __restrict__ ei, const int* __restrict__ offs,
                          int* __restrict__ cur, int* __restrict__ bucket) {
    int e = blockIdx.x * 256 + threadIdx.x;
    if (e < EE) {
        int dst = ei[EE + e];
        int p = atomicAdd(&cur[dst], 1);
        bucket[offs[dst] + p] = e;
    }
}

// ---------------- h [C,N] -> hT [N,C] (LDS tiled transpose) ----------------
__global__ void k_htr(const float* __restrict__ h, float* __restrict__ hT) {
    __shared__ float tile[32][33];
    int n0 = blockIdx.x * 32, c0 = blockIdx.y * 32;
    int tx = threadIdx.x, ty = threadIdx.y;
    tile[ty][tx] = h[(c0 + ty) * NN + n0 + tx];     // coalesced in n
    __syncthreads();
    hT[(n0 + ty) * CC + c0 + tx] = tile[tx][ty];    // coalesced in c
}

// ---------------- edge aggregation: res[k*C+c][dst] = sum_e X[e,k]*h[c,src] ----------------
__global__ __launch_bounds__(CC) void k_agg(const int* __restrict__ ei,
                                            const float* __restrict__ X,
                                            const float* __restrict__ hT,
                                            const int* __restrict__ offs,
                                            const int* __restrict__ bucket,
                                            float* __restrict__ res) {
    int dst = blockIdx.x;
    int c = threadIdx.x;
    int b0 = offs[dst], b1 = offs[dst + 1];
    float acc[KK];
#pragma unroll
    for (int k = 0; k < KK; ++k) acc[k] = 0.f;
    for (int b = b0; b < b1; ++b) {
        int e = bucket[b];          // uniform across block -> scalar loads
        int src = ei[e];            // uniform
        float hv = hT[src * CC + c];// coalesced 1KB column
#pragma unroll
        for (int k = 0; k < KK; ++k) acc[k] += X[e * KK + k] * hv; // X uniform
    }
#pragma unroll
    for (int k = 0; k < KK; ++k) res[(size_t)(k * CC + c) * NN + dst] = acc[k];
}

// ---------------- dense contraction with fp32 WMMA ----------------
// out[i,n] = bias[i] + sum_{kk<2048} W'[i,kk] * res[kk,n]
// W'[i, kk] = weight[k*C*C + i*C + o], kk = k*C + o
// Per wave: 2 n-tiles sharing one A fragment (2x reuse of LDS A reads).
// W' chunk staged via GLOBAL_LOAD_ASYNC_TO_LDS_B128 (ASYNCcnt path).
__global__ __launch_bounds__(256) void k_gemm(const float* __restrict__ wgt,
                                              const float* __restrict__ res,
                                              const float* __restrict__ bias,
                                              float* __restrict__ out) {
    __shared__ float As[16 * LDSTRIDE];
    const int t = threadIdx.x;
    const int i0 = blockIdx.y * 16;
    const int wave = t >> 5;
    const int lane = t & 31;
    const int n0 = (blockIdx.x * 16 + wave * 2) * 16; // two 16-wide n-tiles per wave
    const int mrow = lane & 15;                       // A fragment M row
    const int kb   = (lane >> 4) * 2;                 // lane half -> K base {0,2}
    const int ncol = lane & 15;                       // B/D column

    v8f acc0 = {0.f, 0.f, 0.f, 0.f, 0.f, 0.f, 0.f, 0.f};
    v8f acc1 = {0.f, 0.f, 0.f, 0.f, 0.f, 0.f, 0.f, 0.f};

    for (int kc = 0; kc < KTOT; kc += KCHUNK) {
        // Async-stage W' chunk: 16 rows x 512 floats = 2048 float4s, 8 per thread.
        // Global runs are contiguous & 16B aligned; LDS offsets (r*516+j)*4 are 16B aligned.
        // The VDST operand is the low 32 bits of the flat shared pointer (== wave-relative
        // LDS byte offset per the aperture rule); deriving it from &As[...] also makes As
        // escape so the compiler must honor the asm as an LDS writer.
#pragma unroll
        for (int it = 0; it < 8; ++it) {
            int idx = t + 256 * it;       // float4 index
            int r   = idx >> 7;           // row 0..15
            int j   = (idx & 127) << 2;   // col 0..508, step 4
            int kk  = kc + j;
            const float* gp = &wgt[(size_t)(kk >> 8) * (CC * CC) +
                                   (size_t)(i0 + r) * CC + (kk & 255)];
            float* lp = &As[r * LDSTRIDE + j];
            unsigned ldsoff = (unsigned)(uintptr_t)lp;
            asm volatile("global_load_async_to_lds_b128 %0, %1, off"
                         :: "v"(ldsoff), "v"(gp)
                         : "memory");
        }
#if __has_builtin(__builtin_amdgcn_s_wait_asynccnt)
        __builtin_amdgcn_s_wait_asynccnt(0);
#else
        asm volatile("s_wait_asynccnt 0" ::: "memory");
#endif
        __syncthreads();

#pragma unroll 4
        for (int kl = 0; kl < KCHUNK; kl += 4) {
            // A fragment (16x4): V0 = K{0|2}, V1 = K{1|3} per lane half; 8B-aligned LDS read
            v2f a = *(const v2f*)&As[mrow * LDSTRIDE + kl + kb];
            int kkg = kc + kl + kb;
            const float* bp = &res[(size_t)kkg * NN + ncol];
            v2f b0, b1;
            b0.x = bp[n0];          b0.y = bp[NN + n0];
            b1.x = bp[n0 + 16];     b1.y = bp[NN + n0 + 16];
            acc0 = __builtin_amdgcn_wmma_f32_16x16x4_f32(
                false, a, false, b0, (short)0, acc0, false, false);
            acc1 = __builtin_amdgcn_wmma_f32_16x16x4_f32(
                false, a, false, b1, (short)0, acc1, false, false);
        }
        __syncthreads();
    }
    // D tile: VGPR r, lane L -> i = i0 + r + 8*(L>>4), n = n0 + (L&15)
    const int ihalf = (lane >> 4) * 8;
#pragma unroll
    for (int r = 0; r < 8; ++r) {
        int i = i0 + r + ihalf;
        float bv = bias[i];
        out[(size_t)i * NN + n0 + ncol]      = acc0[r] + bv;
        out[(size_t)i * NN + n0 + 16 + ncol] = acc1[r] + bv;
    }
}

extern "C" void kernel_launch(void* const* d_in, const int* in_sizes, int n_in,
                              void* d_out, int out_size, void* d_ws, size_t ws_size,
                              hipStream_t stream) {
    const float* h    = (const float*)d_in[0];   // [C, N]
    const float* X    = (const float*)d_in[1];   // [E, K]
    const int*   ei   = (const int*)d_in[2];     // [2, E] (src row, dst row)
    /* d_in[3] = batch_node, unused */
    const float* wgt  = (const float*)d_in[4];   // [K, C, C]
    const float* bias = (const float*)d_in[5];   // [C]
    float* out = (float*)d_out;                  // [C, N]

    char* ws = (char*)d_ws;
    size_t off = 0;
    auto carve = [&](size_t bytes) -> char* {
        char* p = ws + off;
        off += (bytes + 255) & ~size_t(255);
        return p;
    };
    float* hT     = (float*)carve((size_t)NN * CC * 4);       //  4 MB
    float* res    = (float*)carve((size_t)KTOT * NN * 4);     // 32 MB
    int*   cnt    = (int*)carve((size_t)NN * 4);
    int*   offs   = (int*)carve((size_t)(NN + 1) * 4);
    int*   cur    = (int*)carve((size_t)NN * 4);
    int*   bucket = (int*)carve((size_t)EE * 4);

    hipLaunchKernelGGL(k_zero,    dim3(NN / 256), dim3(256), 0, stream, cnt, cur);
    hipLaunchKernelGGL(k_hist,    dim3(EE / 256), dim3(256), 0, stream, ei, cnt);
    hipLaunchKernelGGL(k_scan,    dim3(1),        dim3(32),  0, stream, cnt, offs);
    hipLaunchKernelGGL(k_scatter, dim3(EE / 256), dim3(256), 0, stream, ei, offs, cur, bucket);
    hipLaunchKernelGGL(k_htr,     dim3(NN / 32, CC / 32), dim3(32, 32), 0, stream, h, hT);
    hipLaunchKernelGGL(k_agg,     dim3(NN),       dim3(CC),  0, stream, ei, X, hT, offs, bucket, res);
    hipLaunchKernelGGL(k_gemm,    dim3(16, 16),   dim3(256), 0, stream, wgt, res, bias, out);
}